// Embedder_2551210573866
// MI455X (gfx1250) — compile-verified
//
#include <hip/hip_runtime.h>

// Problem constants (match reference)
#define N_NODES 100000
#define N_EDGES 3200000
#define N_REL   20
#define D_INPUT 16
#define D_HID   32
#define N_POCKET 20000
#define EPS_BN  1e-5f

typedef __attribute__((ext_vector_type(2))) float v2f;
typedef __attribute__((ext_vector_type(8))) float v8f;

// ---------------------------------------------------------------------------
// Pack per-relation weights + self-loop weight into one [K,32] row-major
// matrix (K = (R+1)*d_in).  W is [R, d_in, H] contiguous == [R*d_in, H].
// ---------------------------------------------------------------------------
__global__ void pack_weights_kernel(const float* __restrict__ W,
                                    const float* __restrict__ loopW,
                                    float* __restrict__ wcat, int d_in) {
    int i = blockIdx.x * blockDim.x + threadIdx.x;
    int K = (N_REL + 1) * d_in;
    if (i >= K * D_HID) return;
    int k = i >> 5, o = i & 31;
    float v;
    if (k < N_REL * d_in) v = W[(size_t)k * D_HID + o];
    else                  v = loopW[(size_t)(k - N_REL * d_in) * D_HID + o];
    wcat[i] = v;
}

// ---------------------------------------------------------------------------
// Edge scatter: agg[(dst*R + etype), :] += h[src, :]
// DIN/4 threads per edge, each moving a float4 via hardware f32 atomics.
// ---------------------------------------------------------------------------
template <int DIN>
__global__ void edge_scatter_kernel(const float* __restrict__ h,
                                    const int* __restrict__ src,
                                    const int* __restrict__ dst,
                                    const int* __restrict__ et,
                                    float* __restrict__ agg) {
    constexpr int TPE = DIN / 4;
    long long tid = (long long)blockIdx.x * blockDim.x + threadIdx.x;
    long long e = tid / TPE;
    int t = (int)(tid % TPE);
    if (e >= N_EDGES) return;
    int s = src[e], d = dst[e], r = et[e];
    const float4* hv = (const float4*)(h + (size_t)s * DIN);
    float4 v = hv[t];
    float* out = agg + ((size_t)d * N_REL + r) * DIN + t * 4;
    unsafeAtomicAdd(out + 0, v.x);
    unsafeAtomicAdd(out + 1, v.y);
    unsafeAtomicAdd(out + 2, v.z);
    unsafeAtomicAdd(out + 3, v.w);
}

// ---------------------------------------------------------------------------
// WMMA GEMM: pre[N,32] = [agg | h_in] @ wcat + bias  (optional ReLU),
// with fused per-column BN statistics (sum / sum-of-squares).
// One wave per 16-node tile; two 16x16 output col-tiles per wave.
// Uses V_WMMA_F32_16X16X4_F32; Wcat staged in LDS.
// ---------------------------------------------------------------------------
template <int DIN, bool RELU>
__global__ void rgcn_gemm_kernel(const float* __restrict__ agg,
                                 const float* __restrict__ h_in,
                                 const float* __restrict__ wcat,
                                 const float* __restrict__ biasrow,
                                 float* __restrict__ pre,
                                 float* __restrict__ gsum,
                                 float* __restrict__ gssq) {
    constexpr int K = (N_REL + 1) * DIN;
    extern __shared__ float lds[];
    float* lsum = lds + K * D_HID;   // [32]
    float* lssq = lsum + D_HID;      // [32]

    int tid = threadIdx.x;
    for (int i = tid; i < K * D_HID; i += blockDim.x) lds[i] = wcat[i];
    if (tid < 2 * D_HID) lsum[tid] = 0.0f;
    __syncthreads();

    int wave = tid >> 5, lane = tid & 31;
    int tile = blockIdx.x * (blockDim.x >> 5) + wave;
    constexpr int NT = N_NODES / 16;   // 6250, N divisible by 16

    if (tile < NT) {                    // wave-uniform: EXEC stays all-ones
        int m = lane & 15;
        int half = lane >> 4;
        int kb = half * 2;

        v8f acc0 = {};
        v8f acc1 = {};

        // K-chunk 1: aggregated neighbor features, row stride R*DIN
        const float* aRow = agg + (size_t)(tile * 16 + m) * (N_REL * DIN);
#pragma unroll 4
        for (int k = 0; k < N_REL * DIN; k += 4) {
            v2f a = *(const v2f*)(aRow + k + kb);
            const float* wk = lds + (k + kb) * D_HID;
            v2f b0, b1;
            b0.x = wk[m];            b0.y = wk[D_HID + m];
            b1.x = wk[16 + m];       b1.y = wk[D_HID + 16 + m];
            acc0 = __builtin_amdgcn_wmma_f32_16x16x4_f32(false, a, false, b0,
                                                         (short)0, acc0, false, false);
            acc1 = __builtin_amdgcn_wmma_f32_16x16x4_f32(false, a, false, b1,
                                                         (short)0, acc1, false, false);
        }
        // K-chunk 2: self-loop term h_in @ loopW, row stride DIN
        const float* hRow = h_in + (size_t)(tile * 16 + m) * DIN;
#pragma unroll
        for (int k = 0; k < DIN; k += 4) {
            v2f a = *(const v2f*)(hRow + k + kb);
            const float* wk = lds + (N_REL * DIN + k + kb) * D_HID;
            v2f b0, b1;
            b0.x = wk[m];            b0.y = wk[D_HID + m];
            b1.x = wk[16 + m];       b1.y = wk[D_HID + 16 + m];
            acc0 = __builtin_amdgcn_wmma_f32_16x16x4_f32(false, a, false, b0,
                                                         (short)0, acc0, false, false);
            acc1 = __builtin_amdgcn_wmma_f32_16x16x4_f32(false, a, false, b1,
                                                         (short)0, acc1, false, false);
        }

        // Epilogue: bias, optional ReLU, store, BN partial stats.
        float b0v = biasrow[m];
        float b1v = biasrow[16 + m];
        float s0 = 0.f, q0 = 0.f, s1 = 0.f, q1 = 0.f;
        int rowbase = tile * 16 + half * 8;   // C VGPR g -> row g + half*8
#pragma unroll
        for (int g = 0; g < 8; ++g) {
            float v0 = acc0[g] + b0v;
            float v1 = acc1[g] + b1v;
            if (RELU) { v0 = fmaxf(v0, 0.f); v1 = fmaxf(v1, 0.f); }
            size_t ro = (size_t)(rowbase + g) * D_HID;
            pre[ro + m]      = v0;
            pre[ro + 16 + m] = v1;
            s0 += v0; q0 += v0 * v0;
            s1 += v1; q1 += v1 * v1;
        }
        atomicAdd(&lsum[m], s0);       atomicAdd(&lssq[m], q0);
        atomicAdd(&lsum[16 + m], s1);  atomicAdd(&lssq[16 + m], q1);
    }
    __syncthreads();
    if (tid < D_HID) {
        unsafeAtomicAdd(&gsum[tid], lsum[tid]);
        unsafeAtomicAdd(&gssq[tid], lssq[tid]);
    }
}

// ---------------------------------------------------------------------------
// BN: finalize per-column scale/shift from global sums.
// ---------------------------------------------------------------------------
__global__ void bn_finalize_kernel(const float* __restrict__ gsum,
                                   const float* __restrict__ gssq,
                                   const float* __restrict__ gamma,
                                   const float* __restrict__ beta,
                                   float* __restrict__ scale,
                                   float* __restrict__ shift) {
    int c = threadIdx.x;
    if (c < D_HID) {
        const float inv_n = 1.0f / (float)N_NODES;
        float mean = gsum[c] * inv_n;
        float var  = gssq[c] * inv_n - mean * mean;
        float sc   = gamma[c] * rsqrtf(var + EPS_BN);
        scale[c] = sc;
        shift[c] = beta[c] - mean * sc;
    }
}

template <bool RELU>
__global__ void bn_apply_kernel(const float* __restrict__ pre,
                                const float* __restrict__ scale,
                                const float* __restrict__ shift,
                                float* __restrict__ out) {
    int i = blockIdx.x * blockDim.x + threadIdx.x;
    if (i >= N_NODES * D_HID) return;
    int c = i & (D_HID - 1);
    float v = pre[i] * scale[c] + shift[c];
    if (RELU) v = fmaxf(v, 0.f);
    out[i] = v;
}

__global__ void gather_pocket_kernel(const float* __restrict__ h,
                                     const int* __restrict__ idx,
                                     float* __restrict__ out) {
    int i = blockIdx.x * blockDim.x + threadIdx.x;
    if (i >= N_POCKET * D_HID) return;
    int p = i >> 5, c = i & 31;
    out[i] = h[(size_t)idx[p] * D_HID + c];
}

// ---------------------------------------------------------------------------
extern "C" void kernel_launch(void* const* d_in, const int* in_sizes, int n_in,
                              void* d_out, int out_size, void* d_ws, size_t ws_size,
                              hipStream_t stream) {
    const float* nt     = (const float*)d_in[0];
    const int*   src    = (const int*)d_in[1];
    const int*   dst    = (const int*)d_in[2];
    const int*   et     = (const int*)d_in[3];
    const int*   pocket = (const int*)d_in[4];
    const float* W0     = (const float*)d_in[5];
    const float* Wl     = (const float*)d_in[6];
    const float* loop0  = (const float*)d_in[7];
    const float* loopl  = (const float*)d_in[8];
    const float* bias   = (const float*)d_in[9];
    const float* gamma  = (const float*)d_in[10];
    const float* beta   = (const float*)d_in[11];

    // Workspace layout (floats): ~282 MB total
    float* ws   = (float*)d_ws;
    float* h_a  = ws;                                      // normalized h [N,32]
    float* h_b  = h_a + (size_t)N_NODES * D_HID;           // pre-BN conv out [N,32]
    float* agg  = h_b + (size_t)N_NODES * D_HID;           // [N, R, 32] max
    float* wcat = agg + (size_t)N_NODES * N_REL * D_HID;   // [(R+1)*32, 32] max
    float* gsum = wcat + (size_t)(N_REL + 1) * D_HID * D_HID;
    float* gssq = gsum + D_HID;
    float* scl  = gssq + D_HID;
    float* shf  = scl + D_HID;

    const int tiles  = N_NODES / 16;
    const int gblocks = (tiles + 7) / 8;     // 8 waves (tiles) per 256-thread block
    const int bn_blocks = (N_NODES * D_HID + 255) / 256;

    // ---------------- Layer 0: IN=16 -> H=32, ReLU in conv ----------------
    {
        constexpr int DIN = D_INPUT;
        constexpr int K = (N_REL + 1) * DIN;
        hipMemsetAsync(agg, 0, (size_t)N_NODES * N_REL * DIN * sizeof(float), stream);
        hipMemsetAsync(gsum, 0, 2 * D_HID * sizeof(float), stream);
        pack_weights_kernel<<<(K * D_HID + 255) / 256, 256, 0, stream>>>(W0, loop0, wcat, DIN);
        long long work = (long long)N_EDGES * (DIN / 4);
        edge_scatter_kernel<DIN><<<(int)((work + 255) / 256), 256, 0, stream>>>(nt, src, dst, et, agg);
        size_t shmem = ((size_t)K * D_HID + 2 * D_HID) * sizeof(float);
        rgcn_gemm_kernel<DIN, true><<<gblocks, 256, shmem, stream>>>(
            agg, nt, wcat, bias + 0 * D_HID, h_b, gsum, gssq);
        bn_finalize_kernel<<<1, 32, 0, stream>>>(gsum, gssq, gamma + 0 * D_HID,
                                                 beta + 0 * D_HID, scl, shf);
        bn_apply_kernel<false><<<bn_blocks, 256, 0, stream>>>(h_b, scl, shf, h_a);
    }

    // ---------------- Layers 1..3: H=32 -> H=32 ----------------
    for (int i = 0; i < 3; ++i) {
        constexpr int DIN = D_HID;
        constexpr int K = (N_REL + 1) * DIN;
        hipMemsetAsync(agg, 0, (size_t)N_NODES * N_REL * DIN * sizeof(float), stream);
        hipMemsetAsync(gsum, 0, 2 * D_HID * sizeof(float), stream);
        pack_weights_kernel<<<(K * D_HID + 255) / 256, 256, 0, stream>>>(
            Wl + (size_t)i * N_REL * D_HID * D_HID,
            loopl + (size_t)i * D_HID * D_HID, wcat, DIN);
        long long work = (long long)N_EDGES * (DIN / 4);
        edge_scatter_kernel<DIN><<<(int)((work + 255) / 256), 256, 0, stream>>>(h_a, src, dst, et, agg);
        size_t shmem = ((size_t)K * D_HID + 2 * D_HID) * sizeof(float);
        if (i < 2)
            rgcn_gemm_kernel<DIN, true><<<gblocks, 256, shmem, stream>>>(
                agg, h_a, wcat, bias + (i + 1) * D_HID, h_b, gsum, gssq);
        else  // final conv layer: no ReLU inside
            rgcn_gemm_kernel<DIN, false><<<gblocks, 256, shmem, stream>>>(
                agg, h_a, wcat, bias + 3 * D_HID, h_b, gsum, gssq);
        bn_finalize_kernel<<<1, 32, 0, stream>>>(gsum, gssq, gamma + (i + 1) * D_HID,
                                                 beta + (i + 1) * D_HID, scl, shf);
        if (i < 2)
            bn_apply_kernel<false><<<bn_blocks, 256, 0, stream>>>(h_b, scl, shf, h_a);
        else  // external ReLU after final BN
            bn_apply_kernel<true><<<bn_blocks, 256, 0, stream>>>(h_b, scl, shf, h_a);
    }

    // ---------------- Pocket gather ----------------
    gather_pocket_kernel<<<(N_POCKET * D_HID + 255) / 256, 256, 0, stream>>>(
        h_a, pocket, (float*)d_out);
}